// MolConv_16793322127443
// MI455X (gfx1250) — compile-verified
//
#include <hip/hip_runtime.h>
#include <hip/hip_bf16.h>

typedef __attribute__((ext_vector_type(2))) float v2f;
typedef __attribute__((ext_vector_type(8))) float v8f;

#define F_IN   256
#define F_OUT  256
#define NBOND  4
#define LDSROW 260   // 256 + 4 pad: row stride % 64 banks == 4 -> conflict-free A-frag reads
#define EPS    1e-5f

// ---------------------------------------------------------------- zero fill
__global__ void zero_kernel(float* __restrict__ p, int n) {
    int i = blockIdx.x * blockDim.x + threadIdx.x;
    if (i < n) p[i] = 0.0f;
}

// ------------------------------------------------- column sum / sum-of-squares
// One thread per column, each block reduces a contiguous chunk of rows
// (fully coalesced 1KB row reads), then atomically folds into ws stats.
__global__ __launch_bounds__(256) void colstats_kernel(
    const float* __restrict__ atom, float* __restrict__ stats,
    int natoms, int chunk)
{
    const int c  = threadIdx.x;           // column 0..255
    int r0 = blockIdx.x * chunk;
    int r1 = r0 + chunk; if (r1 > natoms) r1 = natoms;
    float s = 0.0f, ss = 0.0f;
    for (int r = r0; r < r1; ++r) {
        float x = atom[(size_t)r * F_IN + c];
        s  += x;
        ss += x * x;
    }
    atomicAdd(&stats[c], s);
    atomicAdd(&stats[F_IN + c], ss);
}

// ------------------------------------------ fold stats into scale/shift pair
__global__ void finalize_kernel(const float* __restrict__ gamma,
                                const float* __restrict__ beta,
                                float* __restrict__ stats, float invN)
{
    const int c = threadIdx.x;
    float mean = stats[c] * invN;
    float var  = stats[F_IN + c] * invN - mean * mean;   // biased, torch-BN style
    float sc   = gamma[c] * rsqrtf(var + EPS);
    stats[512 + c] = sc;                                  // scale
    stats[768 + c] = beta[c] - mean * sc;                 // shift
}

// --------------------------------------------------- fused BN+ELU+GEMM (WMMA)
// Block = 16 atom rows x 8 waves. x-tile normalized+ELU'd into LDS once,
// each wave computes 8 of the 64 16x16 output tiles with V_WMMA_F32_16X16X4_F32
// over K=256, and stores directly into the transposed h2[B*N, F_OUT] layout.
__global__ __launch_bounds__(256) void gemm_kernel(
    const float* __restrict__ atom, const float* __restrict__ Wm,
    const float* __restrict__ bias, const float* __restrict__ stats,
    float* __restrict__ h2, int natoms)
{
    __shared__ float xlds[16 * LDSROW];

    const int row0 = blockIdx.x * 16;
    const int tid  = threadIdx.x;

    // ---- stage 16x256 tile: normalize -> ELU -> LDS (coalesced global reads)
    const float* __restrict__ src   = atom + (size_t)row0 * F_IN;
    const float* __restrict__ scale = stats + 512;
    const float* __restrict__ shift = stats + 768;
    for (int i = tid; i < 16 * F_IN; i += 256) {
        const int r = i >> 8;
        const int c = i & 255;
        float x = src[i] * scale[c] + shift[c];
        x = (x > 0.0f) ? x : (__expf(x) - 1.0f);          // ELU(alpha=1)
        xlds[r * LDSROW + c] = x;
    }
    __syncthreads();

    const int lane = tid & 31;
    const int wave = tid >> 5;
    const int mrow = lane & 15;                 // M row within tile
    const int koff = (lane >> 4) << 1;          // 0 for lanes 0-15, 2 for 16-31
    const float* __restrict__ xrow = &xlds[mrow * LDSROW + koff];

    for (int nt = wave; nt < (NBOND * F_OUT) / 16; nt += 8) {
        const int jglob = nt * 16 + mrow;                       // output feature
        const float* __restrict__ wrow = Wm + (size_t)jglob * F_IN + koff;

        v8f acc = {};
        #pragma unroll 8
        for (int k = 0; k < F_IN; k += 4) {
            // A frag: x[M][k+koff], x[M][k+koff+1]   (16x4 f32 layout)
            v2f a; a.x = xrow[k];  a.y = xrow[k + 1];
            // B frag: W[j][k+koff], W[j][k+koff+1]   (B = W^T, 4x16 f32 layout)
            v2f bf; bf.x = wrow[k]; bf.y = wrow[k + 1];
            acc = __builtin_amdgcn_wmma_f32_16x16x4_f32(
                false, a, false, bf, (short)0, acc, false, false);
        }

        const float bj   = bias[jglob];
        const int   bq   = jglob >> 8;          // bond type   (jglob / 256)
        const int   jcol = jglob & 255;         // column within F_OUT
        const int   mb   = (lane >> 4) * 8;     // C/D layout: VGPR r -> M=r / r+8
        float* __restrict__ dst =
            h2 + ((size_t)bq * natoms + row0 + mb) * F_OUT + jcol;
        #pragma unroll
        for (int r = 0; r < 8; ++r)
            dst[(size_t)r * F_OUT] = acc[r] + bj;
    }
}

// ------------------------------------- edge gather * val -> atomic segment sum
// One 256-thread block per edge iteration: lane f handles feature f,
// coalesced 1KB row gather from h2 + GLOBAL_ATOMIC_ADD_F32 into out.
__global__ __launch_bounds__(256) void scatter_kernel(
    const int* __restrict__ erow, const int* __restrict__ ecol,
    const float* __restrict__ eval, const float* __restrict__ h2,
    float* __restrict__ out, int nedges)
{
    const int f = threadIdx.x;
    for (int e = blockIdx.x; e < nedges; e += gridDim.x) {
        const int   r = erow[e];     // uniform per block -> scalar loads
        const int   g = ecol[e];
        const float v = eval[e];
        const float m = v * h2[(size_t)g * F_OUT + f];
        atomicAdd(&out[(size_t)r * F_OUT + f], m);
    }
}

extern "C" void kernel_launch(void* const* d_in, const int* in_sizes, int n_in,
                              void* d_out, int out_size, void* d_ws, size_t ws_size,
                              hipStream_t stream)
{
    const float* atom  = (const float*)d_in[0];
    const float* gamma = (const float*)d_in[1];
    const float* beta  = (const float*)d_in[2];
    const float* Wm    = (const float*)d_in[3];
    const float* bias  = (const float*)d_in[4];
    const int*   erow  = (const int*)d_in[5];
    const int*   ecol  = (const int*)d_in[6];
    const float* eval  = (const float*)d_in[7];
    float* out = (float*)d_out;

    // ws layout (floats): [0,256) colsum, [256,512) colsumsq,
    //                     [512,768) scale, [768,1024) shift, [1024,...) h2
    float* stats = (float*)d_ws;
    float* h2    = stats + 1024;

    const int natoms = in_sizes[0] / F_IN;     // 100000
    const int nedges = in_sizes[5];            // 3200000

    // 1. clear stats accumulators + output
    zero_kernel<<<4, 256, 0, stream>>>(stats, 1024);
    zero_kernel<<<(out_size + 255) / 256, 256, 0, stream>>>(out, out_size);

    // 2. column mean/var partials
    const int chunk  = 256;
    const int sblks  = (natoms + chunk - 1) / chunk;
    colstats_kernel<<<sblks, 256, 0, stream>>>(atom, stats, natoms, chunk);

    // 3. fold into scale/shift
    finalize_kernel<<<1, 256, 0, stream>>>(gamma, beta, stats, 1.0f / (float)natoms);

    // 4. fused BN+ELU+GEMM into transposed h2 (WMMA f32)
    gemm_kernel<<<(natoms + 15) / 16, 256, 0, stream>>>(atom, Wm, bias, stats,
                                                        h2, natoms);

    // 5. sparse gather/scale/segment-sum
    int sg = 131072; if (sg > nedges) sg = nedges;
    scatter_kernel<<<sg, 256, 0, stream>>>(erow, ecol, eval, h2, out, nedges);
}